// BaseContrastHead_43121471652524
// MI455X (gfx1250) — compile-verified
//
#include <hip/hip_runtime.h>
#include <hip/hip_bf16.h>
#include <math.h>

#define NW   8192
#define NS   8192
#define FD   1024
#define EMB  128
#define QN   1024
#define NC   80

typedef __attribute__((ext_vector_type(2))) float v2f;
typedef __attribute__((ext_vector_type(8))) float v8f;

// ---------------------------------------------------------------------------
// Kernel 1: label bookkeeping (counting sort by class) + zero accumulators.
// One block, 128 threads; threads 0..79 each own one class.
// ---------------------------------------------------------------------------
__global__ __launch_bounds__(128)
void setup_labels_kernel(const int* __restrict__ s_lab,
                         const int* __restrict__ w_lab,
                         int* __restrict__ sorted,   // [NS] strong idx sorted by class (stable)
                         int* __restrict__ idx,      // [NW]
                         float* __restrict__ validf, // [NW]
                         float* __restrict__ accum)  // [2]
{
    __shared__ int cs[NC];
    __shared__ int off[NC];
    const int t = threadIdx.x;

    // count strong samples per class
    if (t < NC) {
        int c = 0;
        for (int j = 0; j < NS; ++j) c += (s_lab[j] == t) ? 1 : 0;
        cs[t] = c;
    }
    __syncthreads();

    // serial prefix (80 elems) + zero accumulators
    if (t == 0) {
        int o = 0;
        for (int c = 0; c < NC; ++c) { off[c] = o; o += cs[c]; }
        accum[0] = 0.0f;
        accum[1] = 0.0f;
    }
    __syncthreads();

    // stable counting sort of strong indices by class
    if (t < NC) {
        int k = 0;
        const int o = off[t];
        for (int j = 0; j < NS; ++j) {
            if (s_lab[j] == t) { sorted[o + k] = j; ++k; }
        }
    }
    __syncthreads();

    // per weak sample: occurrence rank within its class -> sel -> idx
    if (t < NC) {
        int r = 0;
        const int c_cs = cs[t];
        const int o = off[t];
        for (int i = 0; i < NW; ++i) {
            if (w_lab[i] == t) {
                if (c_cs > 0) {
                    idx[i] = sorted[o + (r % c_cs)];
                    validf[i] = 1.0f;
                } else {
                    idx[i] = 0;            // argmax of all-false row -> 0 (masked anyway)
                    validf[i] = 0.0f;
                }
                ++r;
            }
        }
    }
}

// ---------------------------------------------------------------------------
// Kernel 2/3: Y[8192x128] = X[8192x1024] @ W[1024x128] + bias
// One wave computes one 16x16 output tile via v_wmma_f32_16x16x4_f32, K=1024.
// Fragment layouts per CDNA5 ISA 7.12.2:
//   A (16x4):  lane L holds M=L%16, a.x=A[M][2*(L/16)], a.y=A[M][2*(L/16)+1]
//   B (4x16):  lane L holds N=L%16, b.x=B[2*(L/16)][N], b.y=B[2*(L/16)+1][N]
//   C/D(16x16): VGPR r, lane L -> row r+8*(L/16), col L%16
// ---------------------------------------------------------------------------
__global__ __launch_bounds__(32)
void gemm_fc_kernel(const float* __restrict__ X,
                    const float* __restrict__ W,
                    const float* __restrict__ bias,
                    float* __restrict__ Y)
{
    const int lane = threadIdx.x;
    const int n = lane & 15;
    const int h = lane >> 4;          // half-wave select
    const int row0 = blockIdx.x * 16;
    const int col0 = blockIdx.y * 16;

    const float* __restrict__ Arow = X + (size_t)(row0 + n) * FD;  // row M=n for this lane

    v8f c = {};
    #pragma unroll 8
    for (int k0 = 0; k0 < FD; k0 += 4) {
        const int ka = k0 + 2 * h;
        v2f a = *(const v2f*)(Arow + ka);                 // 2 consecutive K values (b64)
        v2f b;
        b.x = W[(size_t)ka * EMB + col0 + n];
        b.y = W[(size_t)(ka + 1) * EMB + col0 + n];
        c = __builtin_amdgcn_wmma_f32_16x16x4_f32(
                /*neg_a=*/false, a, /*neg_b=*/false, b,
                /*c_mod=*/(short)0, c, /*reuse_a=*/false, /*reuse_b=*/false);
    }

    const float bn = bias[col0 + n];
    #pragma unroll
    for (int r = 0; r < 8; ++r) {
        Y[(size_t)(row0 + r + 8 * h) * EMB + col0 + n] = c[r] + bn;
    }
}

// ---------------------------------------------------------------------------
// Kernel 4: fused  l_pos (gather dot) + l_neg tiles (WMMA) + online logsumexp
// + masked CE accumulation.  4 waves per block, each wave owns 16 rows.
// l_neg row-block = q[16x128] @ queue_flat[128x1024], 64 N-tiles of 16.
// ---------------------------------------------------------------------------
__global__ __launch_bounds__(128)
void ce_fused_kernel(const float* __restrict__ q,       // [NW x EMB]
                     const float* __restrict__ kmat,    // [NS x EMB]
                     const float* __restrict__ qf,      // queue flat, viewed [EMB x QN]
                     const int* __restrict__ idx,
                     const float* __restrict__ validf,
                     float* __restrict__ accum)
{
    const int lane = threadIdx.x & 31;
    const int wave = threadIdx.x >> 5;                // 0..3
    const int n = lane & 15;
    const int h = lane >> 4;
    const int row0 = (blockIdx.x * 4 + wave) * 16;

    // Preload all A fragments for this 16x128 row block (K=128 -> 32 steps).
    v2f a[32];
    const float* __restrict__ Arow = q + (size_t)(row0 + n) * EMB;
    #pragma unroll
    for (int s = 0; s < 32; ++s) {
        a[s] = *(const v2f*)(Arow + 4 * s + 2 * h);
    }

    // l_pos + running logsumexp state (col 0 of logits is l_pos).
    float lpos = 0.0f, mrun = -1e30f, srun = 0.0f;
    if (lane < 16) {
        const int row = row0 + lane;
        const int ki = idx[row];
        const float* __restrict__ qr = q + (size_t)row * EMB;
        const float* __restrict__ kr = kmat + (size_t)ki * EMB;
        float acc = 0.0f;
        #pragma unroll 8
        for (int e = 0; e < EMB; ++e) acc = fmaf(qr[e], kr[e], acc);
        lpos = acc;
        mrun = acc;        // logits[row][0]
        srun = 1.0f;       // exp(lpos - lpos)
    }

    __shared__ float tile[4][256];   // per-wave 16x16 scratch

    for (int t = 0; t < 64; ++t) {
        v8f c = {};
        const float* __restrict__ Bp = qf + t * 16 + n;
        #pragma unroll
        for (int s = 0; s < 32; ++s) {
            const int ka = 4 * s + 2 * h;
            v2f b;
            b.x = Bp[(size_t)ka * QN];
            b.y = Bp[(size_t)(ka + 1) * QN];
            c = __builtin_amdgcn_wmma_f32_16x16x4_f32(
                    false, a[s], false, b, (short)0, c, false, false);
        }
        #pragma unroll
        for (int r = 0; r < 8; ++r) tile[wave][(r + 8 * h) * 16 + n] = c[r];
        __syncthreads();

        if (lane < 16) {
            const float* tr = &tile[wave][lane * 16];
            float tmax = mrun;
            #pragma unroll
            for (int j = 0; j < 16; ++j) tmax = fmaxf(tmax, tr[j]);
            float s2 = srun * __expf(mrun - tmax);
            #pragma unroll
            for (int j = 0; j < 16; ++j) s2 += __expf(tr[j] - tmax);
            mrun = tmax;
            srun = s2;
        }
        __syncthreads();
    }

    if (lane < 16) {
        const int row = row0 + lane;
        const float ce = (mrun + __logf(srun)) - lpos;   // -log_softmax[:,0]
        const float v = validf[row];
        atomicAdd(&accum[0], ce * v);
        atomicAdd(&accum[1], v);
    }
}

// ---------------------------------------------------------------------------
// Kernel 5: loss = sum_ce / max(count, 1)
// ---------------------------------------------------------------------------
__global__ void finalize_kernel(const float* __restrict__ accum, float* __restrict__ out)
{
    if (threadIdx.x == 0 && blockIdx.x == 0) {
        out[0] = accum[0] / fmaxf(accum[1], 1.0f);
    }
}

// ---------------------------------------------------------------------------
// Host launcher
// ---------------------------------------------------------------------------
extern "C" void kernel_launch(void* const* d_in, const int* in_sizes, int n_in,
                              void* d_out, int out_size, void* d_ws, size_t ws_size,
                              hipStream_t stream)
{
    (void)in_sizes; (void)n_in; (void)out_size; (void)ws_size;

    const float* feats_strong = (const float*)d_in[0];   // [NS, FD]
    const float* feats_weak   = (const float*)d_in[1];   // [NW, FD]
    const int*   strong_lab   = (const int*)d_in[2];     // [NS]
    const int*   weak_lab     = (const int*)d_in[3];     // [NW]
    const float* Wq           = (const float*)d_in[4];   // [FD, EMB]
    const float* bq           = (const float*)d_in[5];   // [EMB]
    const float* Wk           = (const float*)d_in[6];   // [FD, EMB]
    const float* bk           = (const float*)d_in[7];   // [EMB]
    const float* queue        = (const float*)d_in[8];   // [QN, EMB] -> flat [EMB, QN]
    float* out = (float*)d_out;

    // Workspace layout
    float* qv     = (float*)d_ws;                 // NW*EMB
    float* kv     = qv + (size_t)NW * EMB;        // NS*EMB
    int*   sorted = (int*)(kv + (size_t)NS * EMB);// NS
    int*   idx    = sorted + NS;                  // NW
    float* validf = (float*)(idx + NW);           // NW
    float* accum  = validf + NW;                  // 2

    // 1) label bookkeeping + zero accumulators
    hipLaunchKernelGGL(setup_labels_kernel, dim3(1), dim3(128), 0, stream,
                       strong_lab, weak_lab, sorted, idx, validf, accum);

    // 2) q_all = feats_weak @ Wq + bq
    hipLaunchKernelGGL(gemm_fc_kernel, dim3(NW / 16, EMB / 16), dim3(32), 0, stream,
                       feats_weak, Wq, bq, qv);

    // 3) k_all = feats_strong @ Wk + bk
    hipLaunchKernelGGL(gemm_fc_kernel, dim3(NS / 16, EMB / 16), dim3(32), 0, stream,
                       feats_strong, Wk, bk, kv);

    // 4) fused l_pos / l_neg / online logsumexp / masked CE accumulation
    hipLaunchKernelGGL(ce_fused_kernel, dim3(NW / 64), dim3(128), 0, stream,
                       qv, kv, queue, idx, validf, accum);

    // 5) final scalar
    hipLaunchKernelGGL(finalize_kernel, dim3(1), dim3(1), 0, stream, accum, out);
}